// RelativePositionBias2D_34050500723413
// MI455X (gfx1250) — compile-verified
//
#include <hip/hip_runtime.h>
#include <hip/hip_bf16.h>

typedef __attribute__((ext_vector_type(2))) float v2f;
typedef __attribute__((ext_vector_type(4))) float v4f;
typedef __attribute__((ext_vector_type(8))) float v8f;
typedef __attribute__((ext_vector_type(4))) int   v4i;

#define NPOS 3969   // (2*32-1)^2
#define NHEADS 32
#define NSEQ 1024

// ---------------------------------------------------------------------------
// Kernel A: table[3969][32] = relu(coords @ w1 + b1) @ w2  via f32 WMMA.
// One wave (32 threads) per 16-row tile. K = 512, stepped 4 at a time with
// V_WMMA_F32_16X16X4_F32. Two accumulators cover N = 0..15 and 16..31.
//
// ISA layouts (wave32):
//   A 16x4 f32 : lane L holds M = L&15, K = Kbase + (L>>4)*2 + {0,1}
//   B 4x16 f32 : lane L holds N = L&15, K = Kbase + (L>>4)*2 + {0,1}
//   D 16x16 f32: lane L holds N = L&15, rows M = (L>>4)*8 + v, v = 0..7
// ---------------------------------------------------------------------------
__global__ __launch_bounds__(32) void cpb_table_wmma(
    const float* __restrict__ w1,   // [2][512]
    const float* __restrict__ b1,   // [512]
    const float* __restrict__ w2,   // [512][32]
    const float* __restrict__ ct,   // [3969][2] coords table (flattened)
    float* __restrict__ table)      // [3969][32] out
{
  const int lane  = threadIdx.x;            // 0..31, EXEC all ones (no divergence)
  const int m0    = blockIdx.x * 16;
  const int mrow  = m0 + (lane & 15);
  const int mc    = mrow < NPOS ? mrow : (NPOS - 1);   // branchless clamp
  const float c0  = ct[mc * 2 + 0];
  const float c1  = ct[mc * 2 + 1];
  const int khalf = (lane >> 4) * 2;
  const int ncol  = lane & 15;

  v8f acc0 = {};
  v8f acc1 = {};

  for (int kk = 0; kk < 512; kk += 4) {
    const int k0 = kk + khalf;
    // A fragment: hidden activations for this lane's (M, K) pair
    float h0 = fmaf(c0, w1[k0],     fmaf(c1, w1[512 + k0],     b1[k0]));
    float h1 = fmaf(c0, w1[k0 + 1], fmaf(c1, w1[512 + k0 + 1], b1[k0 + 1]));
    v2f a;
    a.x = h0 > 0.0f ? h0 : 0.0f;
    a.y = h1 > 0.0f ? h1 : 0.0f;
    // B fragments: w2 rows k0, k0+1 at columns ncol and ncol+16
    v2f bA, bB;
    bA.x = w2[(k0    ) * NHEADS + ncol];
    bA.y = w2[(k0 + 1) * NHEADS + ncol];
    bB.x = w2[(k0    ) * NHEADS + 16 + ncol];
    bB.y = w2[(k0 + 1) * NHEADS + 16 + ncol];

    acc0 = __builtin_amdgcn_wmma_f32_16x16x4_f32(
        /*neg_a=*/false, a, /*neg_b=*/false, bA,
        /*c_mod=*/(short)0, acc0, /*reuse_a=*/false, /*reuse_b=*/false);
    acc1 = __builtin_amdgcn_wmma_f32_16x16x4_f32(
        false, a, false, bB, (short)0, acc1, false, false);
  }

  const int mbase = m0 + ((lane >> 4) * 8);
#pragma unroll
  for (int v = 0; v < 8; ++v) {
    const int row = mbase + v;
    if (row < NPOS) {
      table[row * NHEADS + ncol]      = acc0[v];
      table[row * NHEADS + 16 + ncol] = acc1[v];
    }
  }
}

// ---------------------------------------------------------------------------
// Kernel B: out[h][i][j] = 16*sigmoid(table[rel_index[i][j]][h]).
// One block per row i (1024 blocks, 256 threads). Each thread owns 4 j's:
//  - v4i coalesced index load
//  - heads in chunks of 4: v4f loads of table[idx*32 + h..h+3]; each idx's
//    32 heads are one 128B line, fully consumed across the h-loop (L0/L2 hit)
//  - v4f nontemporal stores: per wave, 512B contiguous in out[h][i][.]
// ---------------------------------------------------------------------------
__device__ __forceinline__ float sig16(float x) {
  return 16.0f / (1.0f + __expf(-x));
}

__global__ __launch_bounds__(256) void gather_sigmoid(
    const float* __restrict__ table,     // [3969][32]
    const int*   __restrict__ rel,       // [1024][1024]
    float*       __restrict__ out)       // [32][1024][1024]
{
  const int i  = blockIdx.x;             // row 0..1023
  const int j0 = threadIdx.x * 4;        // 4 columns per thread

  const v4i idx = *(const v4i*)(rel + (size_t)i * NSEQ + j0);
  const float* __restrict__ t0 = table + (size_t)idx.x * NHEADS;
  const float* __restrict__ t1 = table + (size_t)idx.y * NHEADS;
  const float* __restrict__ t2 = table + (size_t)idx.z * NHEADS;
  const float* __restrict__ t3 = table + (size_t)idx.w * NHEADS;

  float* __restrict__ obase = out + (size_t)i * NSEQ + j0;

#pragma unroll 2
  for (int hc = 0; hc < NHEADS; hc += 4) {
    const v4f a = *(const v4f*)(t0 + hc);
    const v4f b = *(const v4f*)(t1 + hc);
    const v4f c = *(const v4f*)(t2 + hc);
    const v4f d = *(const v4f*)(t3 + hc);

    v4f r;
    r.x = sig16(a.x); r.y = sig16(b.x); r.z = sig16(c.x); r.w = sig16(d.x);
    __builtin_nontemporal_store(r, (v4f*)(obase + (size_t)(hc + 0) * NSEQ * NSEQ));
    r.x = sig16(a.y); r.y = sig16(b.y); r.z = sig16(c.y); r.w = sig16(d.y);
    __builtin_nontemporal_store(r, (v4f*)(obase + (size_t)(hc + 1) * NSEQ * NSEQ));
    r.x = sig16(a.z); r.y = sig16(b.z); r.z = sig16(c.z); r.w = sig16(d.z);
    __builtin_nontemporal_store(r, (v4f*)(obase + (size_t)(hc + 2) * NSEQ * NSEQ));
    r.x = sig16(a.w); r.y = sig16(b.w); r.z = sig16(c.w); r.w = sig16(d.w);
    __builtin_nontemporal_store(r, (v4f*)(obase + (size_t)(hc + 3) * NSEQ * NSEQ));
  }
}

extern "C" void kernel_launch(void* const* d_in, const int* in_sizes, int n_in,
                              void* d_out, int out_size, void* d_ws, size_t ws_size,
                              hipStream_t stream) {
  // setup_inputs order: w1[2,512], b1[512], w2[512,32], coords_table[1,63,63,2],
  //                     rel_index[1024,1024] (int32)
  const float* w1 = (const float*)d_in[0];
  const float* b1 = (const float*)d_in[1];
  const float* w2 = (const float*)d_in[2];
  const float* ct = (const float*)d_in[3];
  const int*  rel = (const int*)d_in[4];
  float* out   = (float*)d_out;
  float* table = (float*)d_ws;            // 3969*32*4 = ~508 KB scratch

  // Phase 1: MLP -> table via f32 WMMA (249 row tiles of 16)
  cpb_table_wmma<<<(NPOS + 15) / 16, 32, 0, stream>>>(w1, b1, w2, ct, table);
  // Phase 2: bandwidth-bound gather + 16*sigmoid, 128 MiB NT stores
  gather_sigmoid<<<NSEQ, 256, 0, stream>>>(table, rel, out);
}